// MemoryReadHead_84499186581789
// MI455X (gfx1250) — compile-verified
//
#include <hip/hip_runtime.h>
#include <math.h>

// Problem sizes fixed by setup_inputs(): B=32, N=2048, V=64, H=1024, k=16.
#define B_  32
#define N_  2048
#define V_  64
#define H_  1024
#define K_  16
#define EPS_ 1e-8f

typedef __attribute__((ext_vector_type(2))) float v2f;
typedef __attribute__((ext_vector_type(8))) float v8f;

// ---------------------------------------------------------------------------
// Kernel 1: read_key = h @ Wk^T + bk   via V_WMMA_F32_16X16X4_F32 (fp32 WMMA).
// M=32, N=64, K=1024 -> 2x4 tiles of 16x16, one wave per tile, 8 waves = 1 block.
// A (16x4) per-lane layout (ISA 7.12.2): lanes 0-15 hold K={0,1}, lanes 16-31
// hold K={2,3} of row M=lane&15.  B (4x16): VGPR0/1 hold K={0,1} (lanes 0-15)
// and K={2,3} (lanes 16-31) of column N=lane&15; since B[k][n] = Wk[n][k],
// each lane's pair is contiguous in the Wk row -> clean float2 loads.
// C/D (16x16): element r of the v8f = row (r + 8*(lane>=16)), col = lane&15.
// ---------------------------------------------------------------------------
__global__ __launch_bounds__(256) void k_read_key_wmma(
    const float* __restrict__ h,   // (32, 1024)
    const float* __restrict__ Wk,  // (64, 1024)
    const float* __restrict__ bk,  // (64,)
    float* __restrict__ read_key)  // (32, 64)
{
    const int lane = threadIdx.x & 31;
    const int wid  = threadIdx.x >> 5;      // 0..7
    const int m0   = (wid >> 2) * 16;       // 0 or 16
    const int n0   = (wid & 3)  * 16;       // 0,16,32,48
    const int half = lane >> 4;             // 0 or 1
    const int l    = lane & 15;

    const float* __restrict__ hrow = h  + (size_t)(m0 + l) * H_;
    const float* __restrict__ wrow = Wk + (size_t)(n0 + l) * H_;

    v8f c = {};
    for (int kk = 0; kk < H_; kk += 4) {
        const int ko = kk + 2 * half;
        v2f a, b;
        a.x = hrow[ko];     a.y = hrow[ko + 1];   // A[M=l][K=ko..ko+1]
        b.x = wrow[ko];     b.y = wrow[ko + 1];   // B[K=ko..ko+1][N=l]
        c = __builtin_amdgcn_wmma_f32_16x16x4_f32(
                /*neg_a=*/false, a, /*neg_b=*/false, b,
                /*c_mod=*/(short)0, c, /*reuse_a=*/false, /*reuse_b=*/false);
    }

    const float bias = bk[n0 + l];
#pragma unroll
    for (int r = 0; r < 8; ++r) {
        const int row = m0 + r + 8 * half;
        read_key[(size_t)row * V_ + n0 + l] = c[r] + bias;
    }
}

// ---------------------------------------------------------------------------
// Kernel 2: per-batch scalars — read_strength[b] = softplus(h[b]·Ws + bs),
// kn[b] = ||read_key[b]||.  One wave, lane b handles batch b. Negligible cost.
// ---------------------------------------------------------------------------
__global__ void k_scalars(const float* __restrict__ h,
                          const float* __restrict__ Ws,   // (1, 1024)
                          const float* __restrict__ bs,   // (1,)
                          const float* __restrict__ read_key,
                          float* __restrict__ strength,   // (32,)
                          float* __restrict__ kn)         // (32,)
{
    const int b = threadIdx.x;
    if (b >= B_) return;
    const float4* hr  = (const float4*)(h + (size_t)b * H_);
    const float4* ws4 = (const float4*)Ws;
    float acc = 0.f;
#pragma unroll 4
    for (int i = 0; i < H_ / 4; ++i) {
        float4 a = hr[i], w = ws4[i];
        acc += a.x * w.x + a.y * w.y + a.z * w.z + a.w * w.w;
    }
    acc += bs[0];
    strength[b] = (acc > 20.f) ? acc : log1pf(expf(acc));   // stable softplus

    const float4* kr = (const float4*)(read_key + (size_t)b * V_);
    float ss = 0.f;
#pragma unroll
    for (int i = 0; i < V_ / 4; ++i) {
        float4 v = kr[i];
        ss += v.x * v.x + v.y * v.y + v.z * v.z + v.w * v.w;
    }
    kn[b] = sqrtf(ss);
}

// ---------------------------------------------------------------------------
// Kernel 3: the bandwidth-dominant pass. One thread per memory row: stream the
// 64-float row once (16x b128 loads), computing dot(key,row) and ||row||^2 in
// the same pass, then logit = strength * dot / max(kn*mn, eps).
// Grid: (N/256, B). Key row staged in LDS (one 64-float broadcast per block).
// ---------------------------------------------------------------------------
__global__ __launch_bounds__(256) void k_logits(
    const float* __restrict__ mem,       // (32, 2048, 64)
    const float* __restrict__ read_key,  // (32, 64)
    const float* __restrict__ strength,
    const float* __restrict__ kn,
    float* __restrict__ logits)          // (32, 2048)
{
    __shared__ __align__(16) float key[V_];
    const int b = blockIdx.y;
    const int n = blockIdx.x * 256 + threadIdx.x;
    if (threadIdx.x < V_) key[threadIdx.x] = read_key[(size_t)b * V_ + threadIdx.x];
    __syncthreads();

    const float4* __restrict__ row = (const float4*)(mem + ((size_t)b * N_ + n) * V_);
    const float4* __restrict__ k4  = (const float4*)key;
    float dot = 0.f, ss = 0.f;
#pragma unroll
    for (int i = 0; i < V_ / 4; ++i) {
        float4 m = row[i];
        float4 kv = k4[i];
        dot += m.x * kv.x + m.y * kv.y + m.z * kv.z + m.w * kv.w;
        ss  += m.x * m.x  + m.y * m.y  + m.z * m.z  + m.w * m.w;
    }
    const float denom = fmaxf(kn[b] * sqrtf(ss), EPS_);
    logits[(size_t)b * N_ + n] = strength[b] * dot / denom;
}

// ---------------------------------------------------------------------------
// Kernel 4: per-batch top-16 + softmax + outputs. One block per batch.
// Logits (8 KB) live in LDS; 16 iterations of block-wide argmax (selected
// entries knocked to -inf), then serial 16-way softmax, scatter into the
// zeroed weights row, and a 16-row gather for read_vector (threads 0..63).
// ---------------------------------------------------------------------------
__global__ __launch_bounds__(256) void k_topk(
    const float* __restrict__ logits,   // (32, 2048)
    const float* __restrict__ mem,      // (32, 2048, 64)
    float* __restrict__ out_rv,         // (32, 64)
    float* __restrict__ out_w)          // (32, 2048)
{
    __shared__ float sv[N_];
    __shared__ float rv[256];
    __shared__ int   ri[256];
    __shared__ float selv[K_];
    __shared__ int   seli[K_];
    __shared__ float w16[K_];

    const int b = blockIdx.x;
    const int t = threadIdx.x;

    for (int j = t; j < N_; j += 256) {
        sv[j] = logits[(size_t)b * N_ + j];
        out_w[(size_t)b * N_ + j] = 0.f;     // zero the sparse weight row
    }
    __syncthreads();

    for (int it = 0; it < K_; ++it) {
        float best = -INFINITY; int bi = N_;
#pragma unroll
        for (int j = t; j < N_; j += 256) {
            float v = sv[j];
            if (v > best || (v == best && j < bi)) { best = v; bi = j; }
        }
        rv[t] = best; ri[t] = bi;
        __syncthreads();
        for (int s = 128; s > 0; s >>= 1) {
            if (t < s) {
                float v2 = rv[t + s]; int i2 = ri[t + s];
                if (v2 > rv[t] || (v2 == rv[t] && i2 < ri[t])) { rv[t] = v2; ri[t] = i2; }
            }
            __syncthreads();
        }
        if (t == 0) { selv[it] = rv[0]; seli[it] = ri[0]; sv[ri[0]] = -INFINITY; }
        __syncthreads();
    }

    if (t == 0) {
        float mx = selv[0];
        for (int i = 1; i < K_; ++i) mx = fmaxf(mx, selv[i]);
        float s = 0.f;
        for (int i = 0; i < K_; ++i) { w16[i] = expf(selv[i] - mx); s += w16[i]; }
        const float inv = 1.f / s;
        for (int i = 0; i < K_; ++i) w16[i] *= inv;
    }
    __syncthreads();

    if (t < K_) out_w[(size_t)b * N_ + seli[t]] = w16[t];

    if (t < V_) {
        float acc = 0.f;
#pragma unroll
        for (int i = 0; i < K_; ++i)
            acc += w16[i] * mem[((size_t)b * N_ + seli[i]) * V_ + t];
        out_rv[(size_t)b * V_ + t] = acc;
    }
}

// ---------------------------------------------------------------------------
// Inputs (setup_inputs order): h, memory, prev_read_weights(UNUSED),
// link_matrix(UNUSED), Wk, bk, Ws, bs, k(=16).
// Output: read_vector (32*64) then weights (32*2048), concatenated.
// Workspace (floats): read_key[2048] | strength[32] | kn[32] | logits[65536].
// ---------------------------------------------------------------------------
extern "C" void kernel_launch(void* const* d_in, const int* in_sizes, int n_in,
                              void* d_out, int out_size, void* d_ws, size_t ws_size,
                              hipStream_t stream) {
    const float* h      = (const float*)d_in[0];
    const float* memory = (const float*)d_in[1];
    const float* Wk     = (const float*)d_in[4];
    const float* bk     = (const float*)d_in[5];
    const float* Ws     = (const float*)d_in[6];
    const float* bs     = (const float*)d_in[7];

    float* ws       = (float*)d_ws;
    float* read_key = ws;                 // 2048
    float* strength = ws + 2048;          // 32
    float* kn       = ws + 2080;          // 32
    float* logits   = ws + 2112;          // 65536

    float* out_rv = (float*)d_out;        // (32, 64)
    float* out_w  = (float*)d_out + B_ * V_;  // (32, 2048)

    k_read_key_wmma<<<1, 256, 0, stream>>>(h, Wk, bk, read_key);
    k_scalars<<<1, 32, 0, stream>>>(h, Ws, bs, read_key, strength, kn);
    k_logits<<<dim3(N_ / 256, B_), 256, 0, stream>>>(memory, read_key, strength, kn, logits);
    k_topk<<<B_, 256, 0, stream>>>(logits, memory, out_rv, out_w);
}